// TitanMengerBrain_47193100648821
// MI455X (gfx1250) — compile-verified
//
#include <hip/hip_runtime.h>

typedef __attribute__((ext_vector_type(2))) float v2f;
typedef __attribute__((ext_vector_type(8))) float v8f;

#define N_EXP 20
#define HMS 66   // LDS stride for hmid rows (even -> 8B-aligned b64 loads, low bank conflict)

// Wave-local LDS fence: same-wave DS ops complete in order on CDNA5, so a
// dscnt drain + compiler memory barrier is sufficient for cross-lane RAW
// through wave-private LDS (no workgroup barrier needed).
#define WAVE_LDS_FENCE() asm volatile("s_wait_dscnt 0" ::: "memory")

__device__ __forceinline__ v8f wmma_k4(v2f a, v2f b, v8f c) {
    // V_WMMA_F32_16X16X4_F32 : D = A(16x4,f32) * B(4x16,f32) + C(16x16,f32)
    return __builtin_amdgcn_wmma_f32_16x16x4_f32(false, a, false, b, (short)0, c, false, false);
}

__device__ __forceinline__ float fast_silu(float v) {
    // v * sigmoid(v) with hw v_exp_f32 + v_rcp_f32 (avoids IEEE div sequence)
    return v * __builtin_amdgcn_rcpf(1.f + __expf(-v));
}

__global__ void __launch_bounds__(256)
titan_menger_kernel(const float* __restrict__ x,
                    const float* __restrict__ wq,
                    const float* __restrict__ aw,
                    const float* __restrict__ ab,
                    const float* __restrict__ W1,
                    const float* __restrict__ b1,
                    const float* __restrict__ W2,
                    const float* __restrict__ b2,
                    const float* __restrict__ Wg,
                    const float* __restrict__ bg,
                    float* __restrict__ out)
{
    __shared__ __align__(16) float lds_hmid[8][16 * HMS];
    __shared__ __align__(16) float lds_route[8][N_EXP * 16];
    __shared__ __align__(16) float lds_tot[8][16 * 8];

    const int tid  = threadIdx.x;
    const int wave = tid >> 5;
    const int lane = tid & 31;
    const int n    = lane & 15;   // tile column / A-row index
    const int hi   = lane >> 4;   // half-wave: selects K sub-pair / C row offset
    const int rowBase = blockIdx.x * 128 + wave * 16;

    // ---------------- Step 1: quantum router, one row per lane (lanes 0..15) ----------
    if (lane < 16) {
        const int row = rowBase + lane;
        float sum = 0.f;
        #pragma unroll
        for (int i = 0; i < 8; ++i) sum += x[row * 8 + i];
        const float theta = sum * 0.125f;
        const float c  = __cosf(0.5f * theta), s = __sinf(0.5f * theta);
        const float c2 = c * c, s2 = s * s;
        const float c3 = c2 * c, s3 = s2 * s;
        const float c4 = c2 * c2, s4 = s2 * s2;

        // popcount(INV[j]) for the CX-ring permutation state[j] = pre[INV[j]]
        const int PC[16] = {0,2,2,2, 2,4,2,2, 3,1,3,3, 3,1,1,1};
        float sr[16], si[16];
        #pragma unroll
        for (int j = 0; j < 16; ++j) {
            const int p = PC[j];
            sr[j] = (p == 0) ? c4 : (p == 2) ? -c2 * s2 : (p == 4) ? s4 : 0.f;
            si[j] = (p == 1) ? -c3 * s : (p == 3) ? c * s3 : 0.f;
        }
        // RY(w_q[qb]) on qubit qb (bit qb of the state index); real rotation,
        // re/im evolve independently.
        #pragma unroll
        for (int qb = 0; qb < 4; ++qb) {
            const float h  = 0.5f * wq[qb];
            const float cw = __cosf(h), sw = __sinf(h);
            const int   w  = 1 << qb;
            #pragma unroll
            for (int j = 0; j < 16; ++j) {
                if ((j & w) == 0) {
                    const int j1 = j | w;
                    const float r0 = sr[j], r1 = sr[j1];
                    const float i0 = si[j], i1 = si[j1];
                    sr[j]  = cw * r0 - sw * r1;  sr[j1] = sw * r0 + cw * r1;
                    si[j]  = cw * i0 - sw * i1;  si[j1] = sw * i0 + cw * i1;
                }
            }
        }
        const float PAR[16] = {1,-1,-1,1, -1,1,1,-1, -1,1,1,-1, 1,-1,-1,1};
        float ex = 0.f;
        #pragma unroll
        for (int j = 0; j < 16; ++j) ex += PAR[j] * (sr[j] * sr[j] + si[j] * si[j]);

        // softmax( ex * aw[e] + ab[e] ) over 20 experts
        float lg[N_EXP]; float m = -1e30f;
        #pragma unroll
        for (int e = 0; e < N_EXP; ++e) { lg[e] = ex * aw[e] + ab[e]; m = fmaxf(m, lg[e]); }
        float ssum = 0.f;
        #pragma unroll
        for (int e = 0; e < N_EXP; ++e) { lg[e] = __expf(lg[e] - m); ssum += lg[e]; }
        const float inv = __builtin_amdgcn_rcpf(ssum);
        #pragma unroll
        for (int e = 0; e < N_EXP; ++e) lds_route[wave][e * 16 + lane] = lg[e] * inv;
    }

    // ---------------- A-matrix regs for GEMM1: x tile (16x8), K-steps k0=0,4 ----------
    const int rowM = rowBase + n;               // A: M = lane&15
    const v2f a1_0 = *(const v2f*)(x + rowM * 8 + 0 + 2 * hi);
    const v2f a1_1 = *(const v2f*)(x + rowM * 8 + 4 + 2 * hi);

    v8f tot;
    #pragma unroll
    for (int j = 0; j < 8; ++j) tot[j] = 0.f;

    float* __restrict__ hm = lds_hmid[wave];
    const float* __restrict__ rt = lds_route[wave];
    const int nc = n & 7;                       // clamped column for N=8 operands

    WAVE_LDS_FENCE();   // route visible to whole wave before expert loop

    for (int e = 0; e < N_EXP; ++e) {
        // ---- GEMM1: hmid = silu(x @ W1_e + b1_e), 16x64 via 4 N-tiles x 2 K-steps ----
        const float* __restrict__ w1p = W1 + e * 8 * 64;
        #pragma unroll
        for (int t = 0; t < 4; ++t) {
            const int col = t * 16 + n;
            const float bias = b1[e * 64 + col];
            v8f c1;
            #pragma unroll
            for (int j = 0; j < 8; ++j) c1[j] = bias;
            v2f bb0, bb1;                        // B: K = k0 + 2*hi + {0,1}, N = col
            bb0.x = w1p[(0 + 2 * hi) * 64 + col];
            bb0.y = w1p[(1 + 2 * hi) * 64 + col];
            bb1.x = w1p[(4 + 2 * hi) * 64 + col];
            bb1.y = w1p[(5 + 2 * hi) * 64 + col];
            c1 = wmma_k4(a1_0, bb0, c1);
            c1 = wmma_k4(a1_1, bb1, c1);
            // SiLU + stage to LDS (C layout: row = v + 8*hi, col = col)
            #pragma unroll
            for (int v = 0; v < 8; ++v) {
                hm[(v + 8 * hi) * HMS + col] = fast_silu(c1[v]);
            }
        }
        WAVE_LDS_FENCE();   // hmid tile complete (wave-private)

        // ---- GEMM2: eo = hmid @ W2_e + b2_e  (N padded 8->16, K=64 in 16 steps) ----
        const float* __restrict__ w2p = W2 + e * 64 * 8;
        v8f c2;
        {
            const float bias2v = b2[e * 8 + nc];
            const float bias2 = (n < 8) ? bias2v : 0.f;
            #pragma unroll
            for (int j = 0; j < 8; ++j) c2[j] = bias2;
        }
        #pragma unroll
        for (int kk = 0; kk < 16; ++kk) {
            const int k0 = kk * 4;
            // A from LDS: M = n, K = k0 + 2*hi + {0,1}  (8B-aligned b64 load)
            const v2f a2 = *(const v2f*)(hm + n * HMS + k0 + 2 * hi);
            v2f bv;
            const float bx = w2p[(k0 + 2 * hi) * 8 + nc];
            const float by = w2p[(k0 + 2 * hi + 1) * 8 + nc];
            bv.x = (n < 8) ? bx : 0.f;          // cndmask, keeps EXEC all-ones
            bv.y = (n < 8) ? by : 0.f;
            c2 = wmma_k4(a2, bv, c2);
        }
        // weight by route[row][e], accumulate into total
        #pragma unroll
        for (int v = 0; v < 8; ++v) {
            const float r = rt[e * 16 + v + 8 * hi];   // broadcast LDS read
            tot[v] += r * c2[v];
        }
        WAVE_LDS_FENCE();   // allow hm reuse next expert
    }

    // ---------------- GEMM3: out = total @ Wg^T + bg  (K=8, N padded 8->16) ----------
    float* __restrict__ tp = lds_tot[wave];
    if (n < 8) {
        #pragma unroll
        for (int v = 0; v < 8; ++v) tp[(v + 8 * hi) * 8 + n] = tot[v];
    }
    WAVE_LDS_FENCE();

    v8f c3;
    {
        const float bias3v = bg[nc];
        const float bias3 = (n < 8) ? bias3v : 0.f;
        #pragma unroll
        for (int j = 0; j < 8; ++j) c3[j] = bias3;
    }
    #pragma unroll
    for (int kk = 0; kk < 2; ++kk) {
        const int k0 = kk * 4;
        const v2f a3 = *(const v2f*)(tp + n * 8 + k0 + 2 * hi);   // M = n (all 16 rows valid)
        const v2f wgv = *(const v2f*)(Wg + nc * 8 + k0 + 2 * hi); // B[k=d][n=dd] = Wg[dd][d]
        v2f bv;
        bv.x = (n < 8) ? wgv.x : 0.f;
        bv.y = (n < 8) ? wgv.y : 0.f;
        c3 = wmma_k4(a3, bv, c3);
    }
    if (n < 8) {
        #pragma unroll
        for (int v = 0; v < 8; ++v)
            out[(rowBase + v + 8 * hi) * 8 + n] = c3[v];
    }
}

extern "C" void kernel_launch(void* const* d_in, const int* in_sizes, int n_in,
                              void* d_out, int out_size, void* d_ws, size_t ws_size,
                              hipStream_t stream) {
    const float* x   = (const float*)d_in[0];
    const float* wq  = (const float*)d_in[1];
    const float* aw  = (const float*)d_in[2];
    const float* ab  = (const float*)d_in[3];
    const float* W1  = (const float*)d_in[4];
    const float* b1  = (const float*)d_in[5];
    const float* W2  = (const float*)d_in[6];
    const float* b2  = (const float*)d_in[7];
    const float* Wg  = (const float*)d_in[8];
    const float* bg  = (const float*)d_in[9];
    float* out = (float*)d_out;

    const int B = in_sizes[0] / 8;        // 131072
    const int blocks = B / 128;           // 16 rows/wave * 8 waves/block
    titan_menger_kernel<<<blocks, 256, 0, stream>>>(x, wq, aw, ab, W1, b1, W2, b2, Wg, bg, out);
}